// BatchedKNN_61538291417251
// MI455X (gfx1250) — compile-verified
//
#include <hip/hip_runtime.h>
#include <hip/hip_bf16.h>

typedef float v2f __attribute__((ext_vector_type(2)));
typedef float v8f __attribute__((ext_vector_type(8)));

#define N_PTS   8192
#define BATCH   4
#define CHUNK   2048               // ref points staged in LDS per iteration
#define NCHUNK  (N_PTS / CHUNK)    // 4
#define CTILES  (CHUNK / 16)       // 128 column tiles per chunk
#define KNN     16

// One wave32 per 16-query tile. 8 waves / block -> 128 queries per block.
// grid = B * N/128 = 256 blocks.
__global__ __launch_bounds__(256)
void BatchedKNN_kernel(const float* __restrict__ xyz, long long* __restrict__ out)
{
    // (x, y, z, |r|^2) per staged reference point: 32 KB
    __shared__ float refsh[CHUNK * 4];
    // per-wave 16x16 score tile for layout transpose: 8 KB
    __shared__ float dtile[8 * 256];

    const int tid  = threadIdx.x;
    const int lane = tid & 31;
    const int wave = tid >> 5;
    const int b     = blockIdx.x >> 6;                       // batch
    const int qbase = (blockIdx.x & 63) * 128 + wave * 16;   // first query of this wave's tile
    const int row   = lane & 15;
    const int hi    = lane >> 4;
    const int wbase = wave * 256;

    // ---- A fragment: row m of A = (-2qx, -2qy, -2qz, 1)
    // layout (16x4 f32 A): lanes 0-15 hold K=0,1 ; lanes 16-31 hold K=2,3
    const float* qp = xyz + ((size_t)b * N_PTS + (qbase + row)) * 3;
    const float qx = qp[0], qy = qp[1], qz = qp[2];
    v2f afrag;
    if (hi == 0) { afrag[0] = -2.0f * qx; afrag[1] = -2.0f * qy; }
    else         { afrag[0] = -2.0f * qz; afrag[1] = 1.0f; }

    // ---- per-lane sorted top-16 (this lane owns 8 of the 16 columns of each tile
    // for query row `row`; lane pair (L, L+16) merged at the end)
    float bd[KNN];
    int   bidx[KNN];
#pragma unroll
    for (int j = 0; j < KNN; ++j) { bd[j] = __builtin_huge_valf(); bidx[j] = 0; }

    for (int ch = 0; ch < NCHUNK; ++ch) {
        __syncthreads();   // previous chunk's consumers are done with refsh

        // ---- cooperative stage: CHUNK ref points -> LDS as (x,y,z,|r|^2)
#pragma unroll
        for (int i = 0; i < CHUNK / 256; ++i) {
            const int p = tid + i * 256;
            const float* rp = xyz + ((size_t)b * N_PTS + ch * CHUNK + p) * 3;
            const float x = rp[0], y = rp[1], z = rp[2];
            *(float4*)&refsh[p * 4] = make_float4(x, y, z, x * x + y * y + z * z);
        }
        if (ch + 1 < NCHUNK) {  // warm L2/L0 for the next chunk (global_prefetch_b8)
            __builtin_prefetch(xyz + ((size_t)b * N_PTS + (ch + 1) * CHUNK + tid * 8) * 3, 0, 0);
        }
        __syncthreads();

        const int cbase = ch * CHUNK;
        for (int jt = 0; jt < CTILES; ++jt) {
            // ---- B fragment: column n of B = (rx, ry, rz, |r|^2)
            // layout (4x16 f32 B): lanes 0-15 hold K=0,1 ; lanes 16-31 hold K=2,3
            const int n = jt * 16 + row;
            const float* bp = &refsh[n * 4 + hi * 2];   // 8B aligned -> ds_load_b64
            v2f bfrag; bfrag[0] = bp[0]; bfrag[1] = bp[1];

            // ---- 16x16 adjusted-distance tile in one WMMA (C = 0)
            v8f acc = {};
            acc = __builtin_amdgcn_wmma_f32_16x16x4_f32(
                false, afrag, false, bfrag, (short)0, acc, false, false);

            // ---- scatter D (lane = column c, VGPR r = row r + 8*hi) to LDS
#pragma unroll
            for (int r = 0; r < 8; ++r)
                dtile[wbase + (r + 8 * hi) * 16 + row] = acc[r];
            asm volatile("s_wait_dscnt 0x0" ::: "memory");  // cross-lane LDS RAW

            // ---- gather this lane's query row: 8 contiguous scores (2x ds_load_b128)
            const float* tp = &dtile[wbase + row * 16 + hi * 8];
            float cand[8];
#pragma unroll
            for (int t = 0; t < 8; ++t) cand[t] = tp[t];

            const int ibase = cbase + jt * 16 + hi * 8;
#pragma unroll
            for (int t = 0; t < 8; ++t) {
                const float d = cand[t];
                if (d < bd[KNN - 1]) {               // rare after warm-up
                    bd[KNN - 1]   = d;
                    bidx[KNN - 1] = ibase + t;
#pragma unroll
                    for (int s = KNN - 1; s >= 1; --s) {
                        const float d0 = bd[s - 1], d1 = bd[s];
                        const int   i0 = bidx[s - 1], i1 = bidx[s];
                        const bool  sw = d1 < d0;
                        bd[s - 1]   = sw ? d1 : d0;
                        bidx[s - 1] = sw ? i1 : i0;
                        bd[s]       = sw ? d0 : d1;
                        bidx[s]     = sw ? i0 : i1;
                    }
                }
            }
        }
    }

    // ---- final: merge lane pair (hi=0 cols, hi=1 cols) per query row via LDS
    __syncthreads();   // all waves done reading refsh; reuse it as merge scratch
    {
        float* dd = &refsh[wave * 1024 + row * 32 + hi * 16];
        int*   ii = (int*)&refsh[wave * 1024 + 512 + row * 32 + hi * 16];
#pragma unroll
        for (int j = 0; j < KNN; ++j) { dd[j] = bd[j]; ii[j] = bidx[j]; }
    }
    asm volatile("s_wait_dscnt 0x0" ::: "memory");

    if (hi == 0) {
        const float* dA = &refsh[wave * 1024 + row * 32];
        const float* dB = dA + 16;
        const int*   iA = (const int*)&refsh[wave * 1024 + 512 + row * 32];
        const int*   iB = iA + 16;
        long long* op = out + ((size_t)b * N_PTS + qbase + row) * KNN;
        int i = 0, j = 0;
#pragma unroll
        for (int r = 0; r < KNN; ++r) {
            const float da = dA[i], db = dB[j];
            const int   ia = iA[i], ib = iB[j];
            const bool  ta = (da < db) || (da == db && ia < ib);  // stable by index
            op[r] = ta ? (long long)ia : (long long)ib;
            i += ta ? 1 : 0;
            j += ta ? 0 : 1;
        }
    }
}

extern "C" void kernel_launch(void* const* d_in, const int* in_sizes, int n_in,
                              void* d_out, int out_size, void* d_ws, size_t ws_size,
                              hipStream_t stream) {
    const float* xyz = (const float*)d_in[0];   // [4, 8192, 3] f32
    long long*   out = (long long*)d_out;       // [4, 8192, 16] int64
    (void)in_sizes; (void)n_in; (void)out_size; (void)d_ws; (void)ws_size;
    // 256 blocks x 256 threads: 8 wave32 per block, one 16-query WMMA tile per wave
    BatchedKNN_kernel<<<dim3(256), dim3(256), 0, stream>>>(xyz, out);
}